// ConvLRULayer_81716047773941
// MI455X (gfx1250) — compile-verified
//
#include <hip/hip_runtime.h>
#include <hip/hip_bf16.h>
#include <math.h>

typedef __attribute__((ext_vector_type(2))) float v2f;
typedef __attribute__((ext_vector_type(4))) float v4f;
typedef __attribute__((ext_vector_type(8))) float v8f;

namespace {
constexpr int Bb = 2, Ls = 24, Cc = 32, Hh = 128, Ww = 256;
constexpr int HW  = Hh * Ww;                // 32768 pixels per (b,c) plane
constexpr int CHW = Cc * HW;                // elements between successive time slices
constexpr long LCHW = (long)Ls * CHW;       // elements per batch
constexpr int GROUPS_PER_PLANE = HW / 64;   // 512 groups of 64 pixels
constexpr int GPW = 4;                      // 64-pixel groups per wave
}

// ---------------------------------------------------------------------------
// Setup: per-channel Toeplitz matrix T[c][t][s] = k[c][t-s] (0 outside),
// padded to 32x32.  k[c][t] = sum_r gamma * exp(-t*nu) * cos(t*theta).
// ---------------------------------------------------------------------------
__global__ void lru_build_toeplitz(const float* __restrict__ plog,
                                   float* __restrict__ Tmat) {
  const int c = blockIdx.x;
  __shared__ float kk[32];
  const int tid = threadIdx.x;
  if (tid < 32) {
    float sum = 0.0f;
    if (tid < Ls) {
      const float t = (float)tid;
      for (int r = 0; r < 32; ++r) {
        const float nu  = expf(plog[c * 32 + r]);          // params[0][c][r]
        const float th  = expf(plog[1024 + c * 32 + r]);   // params[1][c][r]
        const float lam = expf(-nu);
        const float gamma = sqrtf(fmaxf(1.0f - lam * lam, 1e-12f));
        sum += gamma * expf(-t * nu) * cosf(t * th);
      }
    }
    kk[tid] = sum;   // kk[t>=24] = 0
  }
  __syncthreads();
  for (int idx = tid; idx < 32 * 32; idx += blockDim.x) {
    const int t = idx >> 5, s = idx & 31;
    const int d = t - s;
    Tmat[c * 1024 + idx] = (d >= 0 && d < Ls) ? kk[d] : 0.0f;
  }
}

// ---------------------------------------------------------------------------
// Main: each wave handles GPW groups of 64 contiguous pixels in one (b,c)
// plane.  Four 16-pixel WMMA tiles are interleaved (tile p uses columns
// 4n+p) so that every global access is b128 (256B contiguous per half-wave):
//   - load:  ld0/ld1[j] = x[s0 / s0+1][pix + 4n .. 4n+3]  (12 b128 loads)
//   - B frag tile p, chunk j = { ld0[j][p], ld1[j][p] }
//   - store: pack {acc[p][v]}_p -> y[t][pix + 4n .. 4n+3]  (16 b128 stores)
// D(t x pix) = A(t x s) x B(s x pix) via V_WMMA_F32_16X16X4_F32; row-tile 0
// covers t=0..15 (K-chunks 0..3), row-tile 1 covers t=16..23 (chunks 0..5).
// ---------------------------------------------------------------------------
__global__ void __launch_bounds__(256) lru_conv_wmma(
    const float* __restrict__ x, const float* __restrict__ Tmat,
    float* __restrict__ y) {
  const int lane = threadIdx.x & 31;
  const int wv   = (int)((blockIdx.x * blockDim.x + threadIdx.x) >> 5);
  const int n    = lane & 15;       // N column index within a tile
  const int half = lane >> 4;       // lane half

  const long group0 = (long)wv * GPW;                    // waves stay in-plane
  const int  plane  = (int)(group0 / GROUPS_PER_PLANE);  // b*C + c
  const int  c      = plane & (Cc - 1);
  const int  b      = plane >> 5;
  const long planebase = (long)b * LCHW + (long)c * HW;

  // --- A fragments (per-channel Toeplitz), loaded once per wave ------------
  v2f aT0[4], aT1[6];
  const float* Tc = Tmat + c * 1024;
  #pragma unroll
  for (int j = 0; j < 6; ++j) {
    const int kb = 4 * j + 2 * half;            // K base for this lane half
    if (j < 4) {
      aT0[j][0] = Tc[n * 32 + kb];
      aT0[j][1] = Tc[n * 32 + kb + 1];
    }
    aT1[j][0] = Tc[(16 + n) * 32 + kb];
    aT1[j][1] = Tc[(16 + n) * 32 + kb + 1];
  }

  for (int i = 0; i < GPW; ++i) {
    const long pix = ((group0 + i) & (GROUPS_PER_PLANE - 1)) << 6;
    const float* __restrict__ xp = x + planebase + pix + 4 * n;
    float* __restrict__ yp       = y + planebase + pix + 4 * n;

    // --- 12 x b128 loads: two time slices per K-chunk ----------------------
    v4f ld0[6], ld1[6];
    #pragma unroll
    for (int j = 0; j < 6; ++j) {
      const int s0 = 4 * j + 2 * half;          // s0, s0+1  (< 24 always)
      ld0[j] = *reinterpret_cast<const v4f*>(xp + (long)s0 * CHW);
      ld1[j] = *reinterpret_cast<const v4f*>(xp + (long)(s0 + 1) * CHW);
    }

    // --- matrix core: 4 interleaved tiles, 10 WMMAs each -------------------
    v8f acc0[4], acc1[4];
    #pragma unroll
    for (int p = 0; p < 4; ++p) {
      v8f a0 = {};   // t = 0..15
      v8f a1 = {};   // t = 16..31 (only 16..23 stored)
      #pragma unroll
      for (int j = 0; j < 6; ++j) {
        v2f bf;
        bf[0] = ld0[j][p];
        bf[1] = ld1[j][p];
        if (j < 4)
          a0 = __builtin_amdgcn_wmma_f32_16x16x4_f32(
              false, aT0[j], false, bf, (short)0, a0, false, false);
        a1 = __builtin_amdgcn_wmma_f32_16x16x4_f32(
            false, aT1[j], false, bf, (short)0, a1, false, false);
      }
      acc0[p] = a0;
      acc1[p] = a1;
    }

    // --- 16 x b128 stores: pack the 4 tiles back into contiguous pixels ----
    #pragma unroll
    for (int v = 0; v < 8; ++v) {
      v4f o;
      #pragma unroll
      for (int p = 0; p < 4; ++p) o[p] = acc0[p][v];
      *reinterpret_cast<v4f*>(yp + (long)(v + 8 * half) * CHW) = o;
    }
    if (half == 0) {
      #pragma unroll
      for (int v = 0; v < 8; ++v) {
        v4f o;
        #pragma unroll
        for (int p = 0; p < 4; ++p) o[p] = acc1[p][v];
        *reinterpret_cast<v4f*>(yp + (long)(16 + v) * CHW) = o;
      }
    }
  }
}

extern "C" void kernel_launch(void* const* d_in, const int* in_sizes, int n_in,
                              void* d_out, int out_size, void* d_ws, size_t ws_size,
                              hipStream_t stream) {
  const float* x    = (const float*)d_in[0];   // (B, L, C, H, W) fp32
  const float* plog = (const float*)d_in[1];   // (2, C, R) fp32
  float* y    = (float*)d_out;                 // (B, L, C, H, W) fp32
  float* Tmat = (float*)d_ws;                  // C * 32 * 32 floats = 128 KB

  lru_build_toeplitz<<<Cc, 256, 0, stream>>>(plog, Tmat);

  const int total_waves = (Bb * Cc * GROUPS_PER_PLANE) / GPW;  // 8192
  const int threads = 256;                                     // 8 waves/block
  const int blocks  = total_waves / (threads / 32);            // 1024
  lru_conv_wmma<<<blocks, threads, 0, stream>>>(x, Tmat, y);

  (void)in_sizes; (void)n_in; (void)out_size; (void)ws_size;
}